// Router_33294586479137
// MI455X (gfx1250) — compile-verified
//
#include <hip/hip_runtime.h>
#include <math.h>

// Router: scores = x @ W^T ([16384,2048] x [64,2048]^T), softmax over 64
// experts, top-2 (indices + probabilities).
//
// MI455X reasoning: x is 134 MB streamed once -> ~5.8us HBM floor @ 23.3 TB/s;
// W (512 KB) stays L2-resident. Bandwidth-bound, so the kernel is structured
// as a continuous HBM->LDS async stream (ASYNCcnt double buffering) feeding
// V_WMMA_F32_16X16X4_F32 (exact fp32 semantics, matching the fp32 reference
// so top-k decisions don't flip from downconversion).
//
// Pipeline per chunk i:
//   issue chunk i+1 async loads into buf[(i+1)&1]   (8 per wave)
//   s_wait_asynccnt 8   -> chunk i's loads landed (async loads complete
//                          in order), chunk i+1 still streaming
//   barrier; WMMA from buf[i&1]; barrier

#define DIM   2048
#define NEXP  64
#define KC    32          // K chunk staged in LDS per iteration
#define NCHUNK (DIM / KC) // 64
#define XSTR  36          // LDS row stride (floats): even + 16B-aligned rows
#define WSTR  36
#define SSTR  65          // score row stride: odd -> conflict-free column scan
#define TPB   64          // tokens per block
#define NWAVE 4           // 128 threads = 4 wave32

typedef __attribute__((ext_vector_type(2))) float v2f;
typedef __attribute__((ext_vector_type(8))) float v8f;

__device__ __forceinline__ void async_b128_to_lds(const float* gaddr, unsigned ldsoff) {
    // CDNA5: GLOBAL_LOAD_ASYNC_TO_LDS_B128 (GV mode, 64-bit addr in VGPR pair,
    // LDS byte offset in VDST VGPR). Tracked by ASYNCcnt.
    asm volatile("global_load_async_to_lds_b128 %0, %1, off"
                 :: "v"(ldsoff), "v"(gaddr) : "memory");
}

__global__ __launch_bounds__(128) void router_kernel(
    const float* __restrict__ x, const float* __restrict__ W,
    int* __restrict__ topi, float* __restrict__ topv, int tokens)
{
    __shared__ float xs[2][TPB * XSTR];        // 2 x  9216 B
    __shared__ float ws[2][NEXP * WSTR];       // 2 x  9216 B
    __shared__ float ss[NWAVE * 16 * SSTR];    //     16640 B  (total 53504 B)

    const int tid  = threadIdx.x;
    const int wave = tid >> 5;
    const int lane = tid & 31;
    const int m    = lane & 15;   // M (A/C row) or N (B/expert col) index
    const int hi   = lane >> 4;   // selects K pair {0,1} vs {2,3}
    const int tokBase = blockIdx.x * TPB;

    // ---- per-thread staging slots (4 x-float4 + 4 w-float4 per chunk) ----
    // slot idx = tid + p*128 over 512 float4 slots; 8 float4 per 32-float row.
    const float* gx[4];
    const float* gw[4];
    unsigned lx[4], lw[4];
#pragma unroll
    for (int p = 0; p < 4; ++p) {
        int idx = tid + p * 128;
        int row = idx >> 3;
        int c4  = idx & 7;
        int rg  = tokBase + row; if (rg >= tokens) rg = tokens - 1;
        gx[p] = x + (size_t)rg * DIM + c4 * 4;
        gw[p] = W + (size_t)row * DIM + c4 * 4;
        lx[p] = (unsigned)(size_t)(&xs[0][row * XSTR + c4 * 4]);
        lw[p] = (unsigned)(size_t)(&ws[0][row * WSTR + c4 * 4]);
    }
    const unsigned xbufB = (unsigned)(TPB * XSTR * sizeof(float));
    const unsigned wbufB = (unsigned)(NEXP * WSTR * sizeof(float));

    v8f acc0 = {}, acc1 = {}, acc2 = {}, acc3 = {};

    // ---- prologue: stream chunk 0 into buffer 0 ----
#pragma unroll
    for (int p = 0; p < 4; ++p) {
        async_b128_to_lds(gx[p], lx[p]);
        async_b128_to_lds(gw[p], lw[p]);
    }

    for (int chunk = 0; chunk < NCHUNK; ++chunk) {
        const int buf = chunk & 1;

        if (chunk + 1 < NCHUNK) {
            // ---- stream chunk i+1 into the alternate buffer ----
            const int      nk   = (chunk + 1) * KC;
            const unsigned nbuf = (chunk + 1) & 1;
#pragma unroll
            for (int p = 0; p < 4; ++p) {
                async_b128_to_lds(gx[p] + nk, lx[p] + nbuf * xbufB);
                async_b128_to_lds(gw[p] + nk, lw[p] + nbuf * wbufB);
            }
            // chunk i's 8 loads done; chunk i+1's 8 still in flight
            asm volatile("s_wait_asynccnt 8" ::: "memory");
        } else {
            asm volatile("s_wait_asynccnt 0" ::: "memory");
        }
        __syncthreads();

        // ---- WMMA over the chunk: 8 k-steps x 4 expert tiles ----
        const float* xrow = &xs[buf][(wave * 16 + m) * XSTR + 2 * hi];
        const float* w0   = &ws[buf][( 0 + m) * WSTR + 2 * hi];
        const float* w1   = &ws[buf][(16 + m) * WSTR + 2 * hi];
        const float* w2   = &ws[buf][(32 + m) * WSTR + 2 * hi];
        const float* w3   = &ws[buf][(48 + m) * WSTR + 2 * hi];
#pragma unroll
        for (int kk = 0; kk < KC; kk += 4) {
            v2f a  = *(const v2f*)(xrow + kk);
            v2f b0 = *(const v2f*)(w0 + kk);
            v2f b1 = *(const v2f*)(w1 + kk);
            v2f b2 = *(const v2f*)(w2 + kk);
            v2f b3 = *(const v2f*)(w3 + kk);
            acc0 = __builtin_amdgcn_wmma_f32_16x16x4_f32(false, a, false, b0, (short)0, acc0, false, false);
            acc1 = __builtin_amdgcn_wmma_f32_16x16x4_f32(false, a, false, b1, (short)0, acc1, false, false);
            acc2 = __builtin_amdgcn_wmma_f32_16x16x4_f32(false, a, false, b2, (short)0, acc2, false, false);
            acc3 = __builtin_amdgcn_wmma_f32_16x16x4_f32(false, a, false, b3, (short)0, acc3, false, false);
        }
        __syncthreads();   // buffer reusable: iteration i+1 streams chunk i+2 into buf[i&1]
    }

    // ---- scores -> LDS (C layout: VGPR r, lanes 0-15 = M=r, 16-31 = M=r+8) ----
    float* srow = &ss[wave * 16 * SSTR];
#pragma unroll
    for (int r = 0; r < 8; ++r) {
        int tr = r + 8 * hi;
        srow[tr * SSTR +  0 + m] = acc0[r];
        srow[tr * SSTR + 16 + m] = acc1[r];
        srow[tr * SSTR + 32 + m] = acc2[r];
        srow[tr * SSTR + 48 + m] = acc3[r];
    }
    __syncthreads();

    // ---- per-token softmax + top-2 (16 lanes per wave, one token each) ----
    if (lane < 16) {
        const float* s = &ss[(wave * 16 + m) * SSTR];
        float m1 = -3.0e38f, m2 = -3.0e38f;
        int   i1 = 0, i2 = 0;
        for (int e = 0; e < NEXP; ++e) {
            float v = s[e];
            if (v > m1)      { m2 = m1; i2 = i1; m1 = v; i1 = e; }
            else if (v > m2) { m2 = v;  i2 = e; }
        }
        float sum = 0.f;
        for (int e = 0; e < NEXP; ++e) sum += __expf(s[e] - m1);
        float inv = 1.f / sum;
        int tok = tokBase + wave * 16 + m;
        if (tok < tokens) {
            topi[tok * 2 + 0] = i1;
            topi[tok * 2 + 1] = i2;
            topv[tok * 2 + 0] = inv;                    // exp(0)/sum
            topv[tok * 2 + 1] = __expf(m2 - m1) * inv;
        }
    }
}

extern "C" void kernel_launch(void* const* d_in, const int* in_sizes, int n_in,
                              void* d_out, int out_size, void* d_ws, size_t ws_size,
                              hipStream_t stream) {
    const float* x = (const float*)d_in[0];
    const float* W = (const float*)d_in[1];
    int tokens = in_sizes[0] / DIM;                 // 4*4096 = 16384
    int*   topi = (int*)d_out;                      // first output, int32 bits
    float* topv = (float*)d_out + (size_t)tokens * 2;
    int blocks = (tokens + TPB - 1) / TPB;
    router_kernel<<<blocks, 128, 0, stream>>>(x, W, topi, topv, tokens);
}